// GaussianImage_79912161509595
// MI455X (gfx1250) — compile-verified
//
#include <hip/hip_runtime.h>
#include <hip/hip_bf16.h>

#define NG 1500
#define HPLANE 256
#define PFLOATS (NG * 8)      // 12000 floats = 48000 bytes of params

typedef __attribute__((ext_vector_type(2))) float    v2f;
typedef __attribute__((ext_vector_type(8))) float    v8f;
typedef __attribute__((ext_vector_type(4))) unsigned v4u;
typedef __attribute__((ext_vector_type(8))) int      v8i;
typedef __attribute__((ext_vector_type(4))) int      v4i;

#define K2 0.7213475204444817f   // 0.5*log2(e)

// ---------------------------------------------------------------------------
// Kernel 1: per-gaussian preprocessing. Inverse covariance is PRE-SCALED by
// -K2 so the render loop's exp2 argument is a plain fma chain:
//   params[n] = { -K2*inv_a, -2*K2*inv_b, -K2*inv_c, mx, my, r, g, b }
// color already includes sigmoid(alpha); rgb_a = tanh(rgb)+0.5
// ---------------------------------------------------------------------------
__global__ void gs_prep(const float* __restrict__ mean,
                        const float* __restrict__ alpha,
                        const float* __restrict__ scale,
                        const float* __restrict__ theta,
                        const float* __restrict__ rgb,
                        float* __restrict__ params,
                        unsigned* __restrict__ qbits) {
    int n = blockIdx.x * blockDim.x + threadIdx.x;
    if (n == 0) *qbits = 0x7F800000u;  // +inf, reset every launch (deterministic)
    if (n >= NG) return;

    float ta = (1.0f / (1.0f + __expf(-theta[n]))) * 6.283185307179586f;
    float c = __cosf(ta), s = __sinf(ta);
    float sx = scale[2 * n], sy = scale[2 * n + 1];
    float sx2 = sx * sx, sy2 = sy * sy;
    float cov00 = c * c * sx2 + s * s * sy2 + 1e-6f;
    float cov01 = c * s * (sx2 - sy2);
    float cov11 = s * s * sx2 + c * c * sy2 + 1e-6f;
    float inv_det = 1.0f / (cov00 * cov11 - cov01 * cov01);
    float ia = cov11 * inv_det;
    float ib = -cov01 * inv_det;
    float ic = cov00 * inv_det;

    float mx = tanhf(mean[2 * n]) * 3.0f;
    float my = tanhf(mean[2 * n + 1]) * 3.0f;

    float sa = 1.0f / (1.0f + __expf(-alpha[n]));
    float r = (tanhf(rgb[3 * n + 0]) + 0.5f) * sa;
    float g = (tanhf(rgb[3 * n + 1]) + 0.5f) * sa;
    float b = (tanhf(rgb[3 * n + 2]) + 0.5f) * sa;

    float* p = params + 8 * n;
    p[0] = -K2 * ia; p[1] = -2.0f * K2 * ib; p[2] = -K2 * ic; p[3] = mx;
    p[4] = my;       p[5] = r;               p[6] = g;        p[7] = b;
}

// ---------------------------------------------------------------------------
// Kernel 2: global min of (K2*quad) over all (n,h,w).  smin = K2*qmin, which
// is exactly the additive constant the render loop needs.  Per (gaussian,row)
// quad is convex in x -> discrete min at floor/ceil of x* = mx - (ib/ia)*dy.
// With scaled params a'=-K2*ia, b'=-2K2*ib:  ib/ia = b'/(2a').
// ---------------------------------------------------------------------------
__global__ void gs_qmin(const float* __restrict__ params,
                        unsigned* __restrict__ qbits) {
    int idx = blockIdx.x * blockDim.x + threadIdx.x;
    float q = 3.0e38f;
    if (idx < NG * HPLANE) {
        int n = idx >> 8;
        int h = idx & 255;
        const float* p = params + 8 * n;
        float a = p[0], b = p[1], c = p[2], mx = p[3], my = p[4];
        float y = -3.0f + 6.0f * (float)h * (1.0f / 255.0f);
        float dy = y - my;
        float xs = mx - (b / (2.0f * a)) * dy;     // continuous row minimizer
        float u = (xs + 3.0f) * (255.0f / 6.0f);   // -> grid units
        int i0 = (int)floorf(u);
        i0 = min(max(i0, 0), 255);
        int i1 = min(i0 + 1, 255);
        float cdy2 = c * dy * dy;
        float dx0 = (-3.0f + 6.0f * (float)i0 * (1.0f / 255.0f)) - mx;
        float dx1 = (-3.0f + 6.0f * (float)i1 * (1.0f / 255.0f)) - mx;
        float q0 = -(a * dx0 * dx0 + b * dx0 * dy + cdy2);   // = K2*quad >= 0
        float q1 = -(a * dx1 * dx1 + b * dx1 * dy + cdy2);
        q = fmaxf(fminf(q0, q1), 0.0f);
    }
    // wave32 min-reduction, then one atomic per wave (uint-bit order valid, q>=0)
    #pragma unroll
    for (int off = 16; off > 0; off >>= 1)
        q = fminf(q, __shfl_xor(q, off, 32));
    if ((threadIdx.x & 31) == 0)
        atomicMin(qbits, __float_as_uint(q));
}

// ---------------------------------------------------------------------------
// Kernel 3: fused splat + normalized-exp + WMMA color accumulation + sigmoid.
// Params table (48KB) is DMA'd to LDS once per block by the Tensor Data Mover
// (tensor_load_to_lds + s_wait_tensorcnt), then the K-loop reads it with LDS
// broadcast loads.  One wave owns 16 contiguous pixels; V_WMMA_F32_16X16X4_F32
// accumulates 4 gaussians per instruction:
//   A (16x4): lane L holds M=L%16, K=2*(L/16)+{0,1}  -> 2 splats/lane
//   B (4x16): lane L holds N=L%16 (color cols 0..2), same K pair
//   C/D:      VGPR v, lane L -> row M=v+8*(L/16), col N=L%16
// splat/max = exp2( smin + a'dx^2 + b'dxdy + c'dy^2 )   (raw v_exp_f32)
// ---------------------------------------------------------------------------
__global__ void __launch_bounds__(256)
gs_render(const float* __restrict__ params,
          const unsigned* __restrict__ qbits,
          float* __restrict__ out) {
    __shared__ float smem[PFLOATS];

    // ---- TDM: async-copy the whole params table global->LDS (wave 0 only) --
    if (threadIdx.x < 32) {
        unsigned long long ga = (unsigned long long)(uintptr_t)params;
        unsigned lds_off = (unsigned)(uintptr_t)&smem[0];   // LDS byte offset
        // D# group 0: count=1 | lds_addr | global_addr[56:0] | type=2
        v4u g0 = { 1u,
                   lds_off,
                   (unsigned)(ga & 0xFFFFFFFFull),
                   (unsigned)((ga >> 32) & 0x01FFFFFFull) | 0x80000000u };
        // D# group 1: data_size=4B; tensor 12000x1; tile 12000x1; stride 12000
        v8i g1 = { (int)(2u << 16),            // data_size=2 (4 bytes)
                   (int)(PFLOATS << 16),       // tensor_dim0[15:0]
                   (int)(1u << 16),            // tensor_dim0 hi=0 | tensor_dim1 lo=1
                   (int)(PFLOATS << 16),       // tensor_dim1 hi=0 | tile_dim0
                   (int)1,                     // tile_dim1=1, tile_dim2=0
                   (int)PFLOATS,               // tensor_dim0_stride lo
                   0, 0 };
        v4i gz4 = { 0, 0, 0, 0 };
        v8i gz8 = { 0, 0, 0, 0, 0, 0, 0, 0 };
        __builtin_amdgcn_tensor_load_to_lds(g0, g1, gz4, gz4, gz8, 0);
        __builtin_amdgcn_s_wait_tensorcnt(0);
    }
    __syncthreads();

    int tid   = blockIdx.x * 256 + threadIdx.x;
    int wave  = tid >> 5;
    int lane  = threadIdx.x & 31;
    int mrow  = lane & 15;              // M index / B column index
    int khalf = lane >> 4;              // upper half-wave handles K+2
    int cm    = min(mrow, 2);           // clamped color channel
    float colmask = (mrow < 3) ? 1.0f : 0.0f;   // loop-invariant B mask

    int p0  = wave * 16;                // first pixel of this wave's M-tile
    int pix = p0 + mrow;                // 16 pixels contiguous in one image row
    int h = pix >> 8, w = pix & 255;
    float px = -3.0f + 6.0f * (float)w * (1.0f / 255.0f);
    float py = -3.0f + 6.0f * (float)h * (1.0f / 255.0f);

    float smin = __uint_as_float(*qbits);   // = K2*qmin, additive in exp2 arg

    v8f acc = {};
    for (int k0 = 0; k0 < NG; k0 += 4) {
        int g0i = k0 + 2 * khalf;
        const float4 pa0 = *(const float4*)&smem[8 * g0i];        // a' b' c' mx
        float my0  = smem[8 * g0i + 4];
        float col0 = smem[8 * g0i + 5 + cm];
        const float4 pa1 = *(const float4*)&smem[8 * g0i + 8];
        float my1  = smem[8 * g0i + 12];
        float col1 = smem[8 * g0i + 13 + cm];

        float dx = px - pa0.w, dy = py - my0;
        float t  = fmaf(pa0.x, dx, pa0.y * dy);
        float e  = fmaf(dx, t, fmaf(pa0.z * dy, dy, smin));
        float s0 = __builtin_amdgcn_exp2f(e);                     // raw v_exp_f32

        dx = px - pa1.w; dy = py - my1;
        t  = fmaf(pa1.x, dx, pa1.y * dy);
        e  = fmaf(dx, t, fmaf(pa1.z * dy, dy, smin));
        float s1 = __builtin_amdgcn_exp2f(e);

        v2f Amat = { s0, s1 };
        v2f Bmat = { col0 * colmask, col1 * colmask };

        acc = __builtin_amdgcn_wmma_f32_16x16x4_f32(
            false, Amat, false, Bmat, (short)0, acc, false, false);
    }

    // Extract live columns (N<3) of the 16x16 accumulator and store sigmoid.
    if (mrow < 3) {
        #pragma unroll
        for (int v = 0; v < 8; v++) {
            int p = p0 + v + 8 * khalf;            // pixel index (row M)
            float e = __builtin_amdgcn_exp2f(-1.4426950408889634f * acc[v]);
            out[p * 3 + mrow] = __builtin_amdgcn_rcpf(1.0f + e);
        }
    }
}

// ---------------------------------------------------------------------------
extern "C" void kernel_launch(void* const* d_in, const int* in_sizes, int n_in,
                              void* d_out, int out_size, void* d_ws, size_t ws_size,
                              hipStream_t stream) {
    const float* mean  = (const float*)d_in[0];  // (N,2,1)
    const float* alpha = (const float*)d_in[1];  // (N,1)
    const float* scale = (const float*)d_in[2];  // (N,2)
    const float* theta = (const float*)d_in[3];  // (N,1)
    const float* rgb   = (const float*)d_in[4];  // (N,3)

    unsigned* qbits  = (unsigned*)d_ws;                   // 4B @ offset 0
    float*    params = (float*)((char*)d_ws + 16);        // NG*8 floats (48KB)
    float*    out    = (float*)d_out;                     // (256,256,3)

    gs_prep<<<(NG + 255) / 256, 256, 0, stream>>>(mean, alpha, scale, theta, rgb,
                                                  params, qbits);
    gs_qmin<<<(NG * HPLANE + 255) / 256, 256, 0, stream>>>(params, qbits);
    // 65536 pixels / 16 per wave = 4096 waves = 512 blocks of 8 waves
    gs_render<<<512, 256, 0, stream>>>(params, qbits, out);
}